// BiMamba_12867722018958
// MI455X (gfx1250) — compile-verified
//
#include <hip/hip_runtime.h>
#include <math.h>

#define L_SEQ     2048
#define DMODEL    512
#define DINNER    1024
#define DSTATE    128
#define HEADDIM   64
#define NHEADSN   16
#define CONVDIM   1280
#define DINPROJ   2320
#define EPS_RMS   1e-5f

typedef __attribute__((ext_vector_type(2))) float v2f;
typedef __attribute__((ext_vector_type(8))) float v8f;

// ---------------------------------------------------------------------------
// GEMM: C[M,N] = A[M,K] (optionally row-flipped) * Bw[N,K]^T   (all fp32)
// One wave computes a 64x16 output block = 4 M-tiles x 1 N-tile.
// B fragment loaded once per K-step, reused by 4 chained
// V_WMMA_F32_16X16X4_F32 ops; fragments software-pipelined one K-step
// ahead (steady-state loop + peeled final step, no per-iter guard).
// Requires M % 64 == 0, N % 16 == 0, K % 4 == 0 (true for all call sites).
// ---------------------------------------------------------------------------
__global__ void bm_wmma_gemm_nt_f32(const float* __restrict__ A,
                                    const float* __restrict__ Bw,
                                    float* __restrict__ C,
                                    int M, int N, int K, int flipA)
{
    const int lane   = threadIdx.x & 31;
    const int wave   = threadIdx.x >> 5;
    const int tileId = blockIdx.x * (blockDim.x >> 5) + wave;
    const int ntn    = N >> 4;
    const int tile_mb = tileId / ntn;       // 64-row block index
    const int tile_n  = tileId - tile_mb * ntn;
    if (tile_mb * 64 >= M) return;          // wave-uniform guard (EXEC all-1s at WMMA)

    const int mn   = lane & 15;             // A: row within tile   B: col N
    const int koff = (lane >> 4) << 1;      // 0 for lanes 0-15, 2 for lanes 16-31

    const float* Ar0;
    const float* Ar1;
    const float* Ar2;
    const float* Ar3;
    {
        int r0 = tile_mb * 64 +  0 + mn;
        int r1 = tile_mb * 64 + 16 + mn;
        int r2 = tile_mb * 64 + 32 + mn;
        int r3 = tile_mb * 64 + 48 + mn;
        if (flipA) { r0 = M - 1 - r0; r1 = M - 1 - r1; r2 = M - 1 - r2; r3 = M - 1 - r3; }
        Ar0 = A + (size_t)r0 * K + koff;
        Ar1 = A + (size_t)r1 * K + koff;
        Ar2 = A + (size_t)r2 * K + koff;
        Ar3 = A + (size_t)r3 * K + koff;
    }
    const float* Br = Bw + (size_t)(tile_n * 16 + mn) * K + koff;

    v8f acc0 = {}, acc1 = {}, acc2 = {}, acc3 = {};

    // current fragments (k = 0)
    v2f b, a0, a1, a2, a3;
    b  = *(const v2f*)(Br);
    a0 = *(const v2f*)(Ar0);
    a1 = *(const v2f*)(Ar1);
    a2 = *(const v2f*)(Ar2);
    a3 = *(const v2f*)(Ar3);

    // steady state: issue loads for step k+4, then 4 WMMAs for step k
    for (int k = 0; k < K - 4; k += 4) {
        if ((k & 63) == 0) {                // gfx1250 global_prefetch_b8
            __builtin_prefetch(Br  + k + 256, 0, 1);
            __builtin_prefetch(Ar0 + k + 256, 0, 1);
            __builtin_prefetch(Ar2 + k + 256, 0, 1);
        }
        const v2f bn  = *(const v2f*)(Br  + k + 4);
        const v2f an0 = *(const v2f*)(Ar0 + k + 4);
        const v2f an1 = *(const v2f*)(Ar1 + k + 4);
        const v2f an2 = *(const v2f*)(Ar2 + k + 4);
        const v2f an3 = *(const v2f*)(Ar3 + k + 4);
        acc0 = __builtin_amdgcn_wmma_f32_16x16x4_f32(false, a0, false, b,
                                                     (short)0, acc0, false, false);
        acc1 = __builtin_amdgcn_wmma_f32_16x16x4_f32(false, a1, false, b,
                                                     (short)0, acc1, false, false);
        acc2 = __builtin_amdgcn_wmma_f32_16x16x4_f32(false, a2, false, b,
                                                     (short)0, acc2, false, false);
        acc3 = __builtin_amdgcn_wmma_f32_16x16x4_f32(false, a3, false, b,
                                                     (short)0, acc3, false, false);
        b = bn; a0 = an0; a1 = an1; a2 = an2; a3 = an3;
    }
    // peeled final step
    acc0 = __builtin_amdgcn_wmma_f32_16x16x4_f32(false, a0, false, b,
                                                 (short)0, acc0, false, false);
    acc1 = __builtin_amdgcn_wmma_f32_16x16x4_f32(false, a1, false, b,
                                                 (short)0, acc1, false, false);
    acc2 = __builtin_amdgcn_wmma_f32_16x16x4_f32(false, a2, false, b,
                                                 (short)0, acc2, false, false);
    acc3 = __builtin_amdgcn_wmma_f32_16x16x4_f32(false, a3, false, b,
                                                 (short)0, acc3, false, false);

    // C/D layout: lane 0-15 -> n=lane, rows M=0..7 in v; lanes 16-31 -> M=8..15
    const int msub = (lane >> 4) << 3;
    float* Cp = C + (size_t)(tile_mb * 64 + msub) * N + tile_n * 16 + mn;
#pragma unroll
    for (int v = 0; v < 8; v++) Cp[(size_t)v * N]              = acc0[v];
#pragma unroll
    for (int v = 0; v < 8; v++) Cp[(size_t)(16 + v) * N]       = acc1[v];
#pragma unroll
    for (int v = 0; v < 8; v++) Cp[(size_t)(32 + v) * N]       = acc2[v];
#pragma unroll
    for (int v = 0; v < 8; v++) Cp[(size_t)(48 + v) * N]       = acc3[v];
}

// ---------------------------------------------------------------------------
// Depthwise causal conv(4) + SiLU on xBC channels; softplus(dt)+dA on heads.
// ---------------------------------------------------------------------------
__global__ void bm_conv_silu_dt(const float* __restrict__ zx,     // [L, DINPROJ]
                                const float* __restrict__ convw,  // [CONVDIM,1,4]
                                const float* __restrict__ convb,  // [CONVDIM]
                                const float* __restrict__ dtb,    // [NHEADS]
                                const float* __restrict__ Alog,   // [NHEADS]
                                float* __restrict__ xbc,          // [L, CONVDIM]
                                float* __restrict__ dt_o,         // [L, NHEADS]
                                float* __restrict__ dA_o)         // [L, NHEADS]
{
    const int idx = blockIdx.x * blockDim.x + threadIdx.x;
    const int W   = CONVDIM + NHEADSN;
    if (idx >= L_SEQ * W) return;
    const int l = idx / W;
    const int c = idx - l * W;
    if (c < CONVDIM) {
        float acc = convb[c];
#pragma unroll
        for (int t = 0; t < 4; t++) {
            const int ll = l - 3 + t;
            if (ll >= 0)
                acc += convw[c * 4 + t] * zx[(size_t)ll * DINPROJ + DINNER + c];
        }
        const float s = acc / (1.0f + expf(-acc));        // SiLU
        xbc[(size_t)l * CONVDIM + c] = s;
    } else {
        const int h = c - CONVDIM;
        const float v  = zx[(size_t)l * DINPROJ + DINNER + CONVDIM + h] + dtb[h];
        const float sp = (v > 20.0f) ? v : log1pf(expf(v));   // softplus
        const float Ah = -expf(Alog[h]);
        dt_o[l * NHEADSN + h] = sp;
        dA_o[l * NHEADSN + h] = expf(sp * Ah);
    }
}

// ---------------------------------------------------------------------------
// Sequential SSM scan, one head per workgroup (16 blocks / direction).
// State h[64 p][128 n] in registers: 256 thr * 32 floats. B/C/x via LDS.
// thread t: p = t>>2, n-chunk = (t&3)*32 .. +31 ; y reduced over lane quads.
// ---------------------------------------------------------------------------
__global__ void bm_ssm_scan(const float* __restrict__ xbc,   // [L, CONVDIM]
                            const float* __restrict__ dt,    // [L, NHEADS]
                            const float* __restrict__ dA,    // [L, NHEADS]
                            float* __restrict__ yscan)       // [L, DINNER]
{
    const int head = blockIdx.x;
    const int tid  = threadIdx.x;
    const int p    = tid >> 2;
    const int nb   = (tid & 3) * 32;

    __shared__ float sx[HEADDIM];
    __shared__ float sB[DSTATE];
    __shared__ float sC[DSTATE];

    float h[32];
#pragma unroll
    for (int j = 0; j < 32; j++) h[j] = 0.0f;

    for (int l = 0; l < L_SEQ; l++) {
        const float* row = xbc + (size_t)l * CONVDIM;
        if (tid < 128) sB[tid]       = row[DINNER + tid];
        else           sC[tid - 128] = row[DINNER + DSTATE + (tid - 128)];
        if (tid < 64)  sx[tid]       = row[head * HEADDIM + tid];
        __syncthreads();

        const float dAl  = dA[l * NHEADSN + head];
        const float coef = dt[l * NHEADSN + head] * sx[p];
        float acc = 0.0f;
#pragma unroll
        for (int j = 0; j < 32; j++) {
            h[j] = h[j] * dAl + coef * sB[nb + j];
            acc += h[j] * sC[nb + j];
        }
        acc += __shfl_xor(acc, 1, 32);
        acc += __shfl_xor(acc, 2, 32);
        if ((tid & 3) == 0)
            yscan[(size_t)l * DINNER + head * HEADDIM + p] = acc;
        __syncthreads();
    }
}

// ---------------------------------------------------------------------------
// y = (yscan + D*x) * silu(z); RMSNorm over DINNER; *normw.  In place.
// One block per sequence position, 256 threads * 4 elems.
// ---------------------------------------------------------------------------
__global__ void bm_gate_rmsnorm(float* __restrict__ y,          // [L, DINNER] in/out
                                const float* __restrict__ xbc,  // [L, CONVDIM]
                                const float* __restrict__ zx,   // [L, DINPROJ]
                                const float* __restrict__ Dp,   // [NHEADS]
                                const float* __restrict__ normw)// [DINNER]
{
    const int l   = blockIdx.x;
    const int tid = threadIdx.x;
    __shared__ float red[8];
    __shared__ float scale;

    float vals[4];
    float ss = 0.0f;
#pragma unroll
    for (int j = 0; j < 4; j++) {
        const int i = tid * 4 + j;
        const int hh = i >> 6;
        float v = y[(size_t)l * DINNER + i] + Dp[hh] * xbc[(size_t)l * CONVDIM + i];
        const float z = zx[(size_t)l * DINPROJ + i];
        v *= z / (1.0f + expf(-z));
        vals[j] = v;
        ss += v * v;
    }
#pragma unroll
    for (int o = 16; o >= 1; o >>= 1) ss += __shfl_xor(ss, o, 32);
    if ((tid & 31) == 0) red[tid >> 5] = ss;
    __syncthreads();
    if (tid == 0) {
        float t = 0.0f;
        for (int i = 0; i < 8; i++) t += red[i];
        scale = rsqrtf(t / (float)DINNER + EPS_RMS);
    }
    __syncthreads();
#pragma unroll
    for (int j = 0; j < 4; j++) {
        const int i = tid * 4 + j;
        y[(size_t)l * DINNER + i] = vals[j] * scale * normw[i];
    }
}

// out = out_f + flip_L(out_b)
__global__ void bm_add_flip(const float* __restrict__ of,
                            const float* __restrict__ ob,
                            float* __restrict__ out)
{
    const int idx = blockIdx.x * blockDim.x + threadIdx.x;
    if (idx >= L_SEQ * DMODEL) return;
    const int l = idx / DMODEL;
    const int j = idx - l * DMODEL;
    out[idx] = of[idx] + ob[(size_t)(L_SEQ - 1 - l) * DMODEL + j];
}

// ---------------------------------------------------------------------------
extern "C" void kernel_launch(void* const* d_in, const int* in_sizes, int n_in,
                              void* d_out, int out_size, void* d_ws, size_t ws_size,
                              hipStream_t stream)
{
    // setup_inputs() order: 8 params x {f,b}, then x.
    const float* Win[2]   = { (const float*)d_in[0],  (const float*)d_in[8]  };
    const float* convw[2] = { (const float*)d_in[1],  (const float*)d_in[9]  };
    const float* convb[2] = { (const float*)d_in[2],  (const float*)d_in[10] };
    const float* dtb[2]   = { (const float*)d_in[3],  (const float*)d_in[11] };
    const float* Alog[2]  = { (const float*)d_in[4],  (const float*)d_in[12] };
    const float* Dp[2]    = { (const float*)d_in[5],  (const float*)d_in[13] };
    const float* normw[2] = { (const float*)d_in[6],  (const float*)d_in[14] };
    const float* Wout[2]  = { (const float*)d_in[7],  (const float*)d_in[15] };
    const float* x        =   (const float*)d_in[16];

    float* ws = (float*)d_ws;
    size_t off = 0;
    float* zx[2];  float* xbc[2]; float* dtw[2]; float* dAw[2];
    float* ysc[2]; float* od[2];
    for (int d = 0; d < 2; d++) { zx[d]  = ws + off; off += (size_t)L_SEQ * DINPROJ; }
    for (int d = 0; d < 2; d++) { xbc[d] = ws + off; off += (size_t)L_SEQ * CONVDIM; }
    for (int d = 0; d < 2; d++) { dtw[d] = ws + off; off += (size_t)L_SEQ * NHEADSN; }
    for (int d = 0; d < 2; d++) { dAw[d] = ws + off; off += (size_t)L_SEQ * NHEADSN; }
    for (int d = 0; d < 2; d++) { ysc[d] = ws + off; off += (size_t)L_SEQ * DINNER;  }
    for (int d = 0; d < 2; d++) { od[d]  = ws + off; off += (size_t)L_SEQ * DMODEL;  }

    const dim3 blk256(256);

    for (int d = 0; d < 2; d++) {
        // 1) in_proj: zx = (flip?) x @ Win^T   M=2048 N=2320 K=512
        {
            const int tiles = (L_SEQ / 64) * (DINPROJ / 16);   // 64-row blocks
            bm_wmma_gemm_nt_f32<<<dim3((tiles + 7) / 8), blk256, 0, stream>>>(
                x, Win[d], zx[d], L_SEQ, DINPROJ, DMODEL, d);
        }
        // 2) conv + SiLU + softplus(dt) + dA
        {
            const int total = L_SEQ * (CONVDIM + NHEADSN);
            bm_conv_silu_dt<<<dim3((total + 255) / 256), blk256, 0, stream>>>(
                zx[d], convw[d], convb[d], dtb[d], Alog[d],
                xbc[d], dtw[d], dAw[d]);
        }
        // 3) sequential scan, one head per WGP
        bm_ssm_scan<<<dim3(NHEADSN), blk256, 0, stream>>>(
            xbc[d], dtw[d], dAw[d], ysc[d]);
        // 4) gate + RMSNorm (in place on ysc)
        bm_gate_rmsnorm<<<dim3(L_SEQ), blk256, 0, stream>>>(
            ysc[d], xbc[d], zx[d], Dp[d], normw[d]);
        // 5) out_proj: od = ysc @ Wout^T   M=2048 N=512 K=1024
        {
            const int tiles = (L_SEQ / 64) * (DMODEL / 16);    // 64-row blocks
            bm_wmma_gemm_nt_f32<<<dim3((tiles + 7) / 8), blk256, 0, stream>>>(
                ysc[d], Wout[d], od[d], L_SEQ, DMODEL, DINNER, 0);
        }
    }
    // 6) out = od[0] + flip(od[1])
    bm_add_flip<<<dim3((L_SEQ * DMODEL + 255) / 256), blk256, 0, stream>>>(
        od[0], od[1], (float*)d_out);
}